// PillarFeatureNetGCN_61065845014852
// MI455X (gfx1250) — compile-verified
//
#include <hip/hip_runtime.h>
#include <hip/hip_bf16.h>

typedef __attribute__((ext_vector_type(2))) float v2f;
typedef __attribute__((ext_vector_type(8))) float v8f;
typedef int v4i __attribute__((vector_size(16)));

#if __has_builtin(__builtin_amdgcn_global_load_async_to_lds_b128)
#define USE_ASYNC_LDS 1
typedef __attribute__((address_space(1))) v4i* gas_v4i_ptr;
typedef __attribute__((address_space(3))) v4i* lds_v4i_ptr;
#else
#define USE_ASYNC_LDS 0
#endif

namespace {
constexpr int kPillars = 150000;
constexpr int kTile    = 16;                 // pillars per (single-wave) block
constexpr int kBlocks  = kPillars / kTile;   // 9375, exact — no tail
}

// One wave (32 lanes) per block, 16 pillars per block.
// Stage 1: async global->LDS copy (ASYNCcnt) of 16 pillars x 32 pts x float4.
// Stage 2: 2-lane-per-pillar masked reduction -> mf[7] per pillar (K-padded to 8).
// Stage 3: 8x V_WMMA_F32_16X16X4_F32 : [16 pillars x 7] @ [7 x 64] in fp32.
// Stage 4: bias + relu + per-pillar floor max(.., sel*relu(b)), store 16x64.
__global__ __launch_bounds__(32)
void pillar_gcn_wmma_kernel(const float* __restrict__ features,   // [N][32][4]
                            const int*   __restrict__ num_points, // [N]
                            const float* __restrict__ Wm,         // [7][64]
                            const float* __restrict__ bias,       // [64]
                            float* __restrict__ out)              // [N][64]
{
  __shared__ float4 stage[544];      // 512 float4 + 1 pad per 16 (bank-conflict-free)
  __shared__ float  mf[kTile][8];    // per-pillar mean feature, mf[*][7] == 0
  __shared__ int    nbuf[kTile];

  const int lane     = threadIdx.x & 31;
  const int tileBase = blockIdx.x * kTile;

  // ---- Stage 1: 16 pillars x 32 float4 into padded LDS ----
  const float4* gf4 = reinterpret_cast<const float4*>(features) + (size_t)tileBase * 32;
#if USE_ASYNC_LDS
  // CDNA5 async copy: memory -> LDS directly, tracked by ASYNCcnt.
#pragma unroll
  for (int j = 0; j < 16; ++j) {
    const int idx = j * 32 + lane;
    __builtin_amdgcn_global_load_async_to_lds_b128(
        (gas_v4i_ptr)(gf4 + idx),
        (lds_v4i_ptr)(&stage[idx + (idx >> 4)]),
        /*offset=*/0, /*cpol=*/0);
  }
#if __has_builtin(__builtin_amdgcn_s_wait_asynccnt)
  __builtin_amdgcn_s_wait_asynccnt(0);
#else
  asm volatile("s_wait_asynccnt 0x0" ::: "memory");
#endif
  __syncthreads();
#else
#pragma unroll
  for (int j = 0; j < 16; ++j) {
    const int idx = j * 32 + lane;
    stage[idx + (idx >> 4)] = gf4[idx];
  }
  __syncthreads();
#endif

  // ---- Stage 2: per-pillar reduction (2 lanes per pillar, 16 points each) ----
  const int pi = lane >> 1;   // local pillar 0..15
  const int h  = lane & 1;    // point half
  const int n  = num_points[tileBase + pi];
  float sv0 = 0.f, sv1 = 0.f, sv2 = 0.f, sv3 = 0.f;  // masked channel sums
  float sa0 = 0.f, sa1 = 0.f, sa2 = 0.f;             // all-32-point coord sums
#pragma unroll
  for (int t = 0; t < 16; ++t) {
    const int idx = lane * 16 + t;                   // == pi*32 + h*16 + t
    const float4 f = stage[idx + (idx >> 4)];
    const float  m = (((h << 4) + t) < n) ? 1.f : 0.f;
    sv0 += m * f.x; sv1 += m * f.y; sv2 += m * f.z; sv3 += m * f.w;
    sa0 += f.x;     sa1 += f.y;     sa2 += f.z;
  }
  sv0 += __shfl_xor(sv0, 1, 32);
  sv1 += __shfl_xor(sv1, 1, 32);
  sv2 += __shfl_xor(sv2, 1, 32);
  sv3 += __shfl_xor(sv3, 1, 32);
  sa0 += __shfl_xor(sa0, 1, 32);
  sa1 += __shfl_xor(sa1, 1, 32);
  sa2 += __shfl_xor(sa2, 1, 32);
  if (h == 0) {
    const float nf  = (float)n;
    const float inv = 1.0f / nf;
    const float pm0 = sa0 * inv, pm1 = sa1 * inv, pm2 = sa2 * inv; // points_mean (all 32 pts / n)
    mf[pi][0] = sv0 * inv;
    mf[pi][1] = sv1 * inv;
    mf[pi][2] = sv2 * inv;
    mf[pi][3] = sv3 * inv;
    mf[pi][4] = (sv0 - nf * pm0) * inv;   // mean of masked f_cluster
    mf[pi][5] = (sv1 - nf * pm1) * inv;
    mf[pi][6] = (sv2 - nf * pm2) * inv;
    mf[pi][7] = 0.f;                      // K=7 -> pad K to 8
    nbuf[pi]  = n;
  }
  __syncthreads();

  // ---- Stage 3: A/B fragments per ISA 16x4 f32 layouts ----
  // A 16x4: lanes 0-15 hold K={0,1} in v0/v1, lanes 16-31 hold K={2,3}.
  const int mrow = lane & 15;
  const int kh   = (lane >> 4) * 2;  // 0 or 2
  v2f a1, a2;
  a1.x = mf[mrow][kh + 0];
  a1.y = mf[mrow][kh + 1];
  a2.x = mf[mrow][4 + kh];           // second K-quad: rows 4..7 (row 7 is the zero pad)
  a2.y = mf[mrow][5 + kh];

  const int mbase = (lane >> 4) * 8; // C/D rows: lanes 0-15 -> M=r, lanes 16-31 -> M=r+8
  float sel[8];                      // 1.0 if pillar has invalid rows (n<32), else 0.0
#pragma unroll
  for (int r = 0; r < 8; ++r) sel[r] = (nbuf[mbase + r] < 32) ? 1.f : 0.f;

#pragma unroll
  for (int nt = 0; nt < 4; ++nt) {
    const int col = nt * 16 + (lane & 15);
    // B 4x16: row K striped across lanes; lanes 0-15 K={0,1}, lanes 16-31 K={2,3}.
    const float w5 = Wm[5 * 64 + col];           // unconditional (in-bounds) load
    v2f b1, b2;
    b1.x = Wm[(kh + 0) * 64 + col];
    b1.y = Wm[(kh + 1) * 64 + col];
    b2.x = Wm[(4 + kh) * 64 + col];
    b2.y = (kh & 2) ? 0.f : w5;                  // W row 7 does not exist -> 0

    v8f acc = {};
    acc = __builtin_amdgcn_wmma_f32_16x16x4_f32(false, a1, false, b1,
                                                (short)0, acc, false, false);
    acc = __builtin_amdgcn_wmma_f32_16x16x4_f32(false, a2, false, b2,
                                                (short)0, acc, false, false);

    // ---- Stage 4: epilogue: max(relu(acc+b), sel * relu(b)) ----
    const float bb = bias[col];
    const float rb = fmaxf(bb, 0.f);             // relu(b): value of invalid rows
#pragma unroll
    for (int r = 0; r < 8; ++r) {
      float v = fmaxf(acc[r] + bb, 0.f);
      v = fmaxf(v, sel[r] * rb);                 // no-op when n==32 (sel==0)
      out[(size_t)(tileBase + mbase + r) * 64 + col] = v;
    }
  }
}

extern "C" void kernel_launch(void* const* d_in, const int* in_sizes, int n_in,
                              void* d_out, int out_size, void* d_ws, size_t ws_size,
                              hipStream_t stream) {
  const float* features   = (const float*)d_in[0];
  const int*   num_points = (const int*)d_in[1];
  // d_in[2] = coors (unused by the reference computation)
  const float* W    = (const float*)d_in[3];
  const float* b    = (const float*)d_in[4];
  float*       outp = (float*)d_out;

  dim3 grid(kBlocks), block(32);
  hipLaunchKernelGGL(pillar_gcn_wmma_kernel, grid, block, 0, stream,
                     features, num_points, W, b, outp);
}